// TopKRouter_9655086482010
// MI455X (gfx1250) — compile-verified
//
#include <hip/hip_runtime.h>
#include <hip/hip_bf16.h>

// ---------------------------------------------------------------------------
// TopK MoE router for MI455X (gfx1250, wave32).
//   logits = input[T,2048] * weight[128,2048]^T   (fp32, V_WMMA_F32_16X16X4_F32)
//   probs, idx = softmax(top8(logits))
// HBM-bound: 134 MB input stream @ 23.3 TB/s ~ 5.8 us floor.
// Staging uses GLOBAL_LOAD_ASYNC_TO_LDS_B128 (ASYNCcnt) with LDS double
// buffering so each workgroup overlaps HBM streaming with WMMA compute.
// ---------------------------------------------------------------------------

typedef __attribute__((ext_vector_type(2))) float v2f;
typedef __attribute__((ext_vector_type(8))) float v8f;

#define HID   2048   // hidden size (K)
#define NEXP  128    // experts (N)
#define TOPK  8
#define MT    64     // tokens per block (M tile)
#define KB    32     // K block staged in LDS
#define NKB   (HID / KB)                 // 64 K blocks
#define AST   36     // A LDS row stride (dwords): mult of 4 (16B align) + conflict-free
#define BST   36     // B LDS row stride (dwords)
#define LSTR  129    // logits LDS row stride (dwords): odd -> conflict-free row reads
#define BOFF  (MT * AST)                 // 2304 dwords: B tile offset within a buffer
#define BUFSZ (MT * AST + NEXP * BST)    // 6912 dwords per double buffer
#define SMEM_DW (2 * BUFSZ)              // 13824 dwords = 55296 B (logits 8256 dw overlap buf0)

typedef __attribute__((address_space(3))) const float lds_cf;

// LDS byte offset (relative to workgroup LDS base) of a pointer into __shared__.
__device__ __forceinline__ unsigned lds_off(const float* p) {
    return (unsigned)(unsigned long long)(lds_cf*)p;
}

// GLOBAL_LOAD_ASYNC_TO_LDS_B128: per-lane 16B HBM -> LDS, no VGPR data, ASYNCcnt.
__device__ __forceinline__ void async_ld16(const float* g, unsigned lds_byte) {
    asm volatile("global_load_async_to_lds_b128 %0, %1, off"
                 :: "v"(lds_byte), "v"((unsigned long long)g)
                 : "memory");
}

__device__ __forceinline__ void wait_async_le12() {
    asm volatile("s_wait_asynccnt 0xc" ::: "memory");
}
__device__ __forceinline__ void wait_async_0() {
    asm volatile("s_wait_asynccnt 0x0" ::: "memory");
}

__global__ __launch_bounds__(128)
void topk_router_kernel(const float* __restrict__ input,
                        const float* __restrict__ weight,
                        float* __restrict__ probs_out,
                        int* __restrict__ idx_out,
                        int num_tokens) {
    __shared__ float smem[SMEM_DW];      // 55 KB (WGP has 320 KB LDS)

    const int tid  = threadIdx.x;        // 0..127
    const int wave = tid >> 5;           // 0..3   (wave32)
    const int lane = tid & 31;
    const int t0   = blockIdx.x * MT;    // first token of this block

    const int lh = lane & 15;            // sub-lane within half-wave
    const int lx = (lane >> 4) * 2;      // K offset: lanes 0-15 -> K0,K1; 16-31 -> K2,K3

    const unsigned smem_base = lds_off(smem);

    v8f acc[8] = {};                     // 8 N-tiles of 16x16 fp32 accumulators

    // ---- stage K-block kb into buffer buf (12 async b128 per thread/wave) ----
    auto stage = [&](int kb, int buf) {
        const int k0 = kb * KB;
        const unsigned base = smem_base + (unsigned)(buf * BUFSZ) * 4u;
        #pragma unroll
        for (int i = 0; i < 4; ++i) {                       // A: 64 rows x 32 K
            int linear = i * 128 + tid;                     // 0..511
            int row = linear >> 3;                          // 8 float4 per row
            int c4  = linear & 7;
            async_ld16(input + (size_t)(t0 + row) * HID + k0 + c4 * 4,
                       base + (unsigned)(row * AST + c4 * 4) * 4u);
        }
        #pragma unroll
        for (int i = 0; i < 8; ++i) {                       // B: 128 experts x 32 K
            int linear = i * 128 + tid;                     // 0..1023
            int row = linear >> 3;
            int c4  = linear & 7;
            async_ld16(weight + (size_t)row * HID + k0 + c4 * 4,
                       base + (unsigned)(BOFF + row * BST + c4 * 4) * 4u);
        }
    };

    stage(0, 0);                         // prologue: fill buffer 0

    for (int kb = 0; kb < NKB; ++kb) {
        const int cur = kb & 1;
        if (kb + 1 < NKB) {
            stage(kb + 1, 1 - cur);      // overlap: next block streams while we compute
            wait_async_le12();           // first 12 (block kb) landed (in-order)
        } else {
            wait_async_0();
        }
        __syncthreads();                 // all waves' tile data visible

        const float* abuf = smem + cur * BUFSZ;
        const float* bbuf = abuf + BOFF;
        #pragma unroll
        for (int kk = 0; kk < KB; kk += 4) {
            // A fragment (16x4): lane lh holds row (16*wave+lh), K = kk+lx, kk+lx+1
            v2f a = *(const v2f*)(abuf + (wave * 16 + lh) * AST + kk + lx);
            #pragma unroll
            for (int n = 0; n < 8; ++n) {
                // B fragment (4x16): lane lh holds expert (16n+lh), same K pair
                v2f b = *(const v2f*)(bbuf + (n * 16 + lh) * BST + kk + lx);
                acc[n] = __builtin_amdgcn_wmma_f32_16x16x4_f32(
                    /*neg_a=*/false, a, /*neg_b=*/false, b,
                    /*c_mod=*/(short)0, acc[n],
                    /*reuse_a=*/false, /*reuse_b=*/false);
            }
        }
        __syncthreads();                 // all waves done reading cur before it is re-filled
    }

    // ---- spill logits to LDS: C layout VGPR j -> row j + 8*(lane>=16), col = lane&15
    const int rbase = wave * 16 + (lane >> 4) * 8;
    #pragma unroll
    for (int n = 0; n < 8; ++n) {
        #pragma unroll
        for (int j = 0; j < 8; ++j) {
            smem[(rbase + j) * LSTR + n * 16 + lh] = acc[n][j];
        }
    }
    __syncthreads();

    // ---- per-token top-8 + softmax (threads 0..63, one token each) ----
    if (tid < MT) {
        const float* row = smem + tid * LSTR;
        float val[TOPK];
        int   idx[TOPK];
        #pragma unroll
        for (int i = 0; i < TOPK; ++i) { val[i] = -3.4e38f; idx[i] = 0; }

        for (int e = 0; e < NEXP; ++e) {
            float v = row[e];
            if (v > val[TOPK - 1]) {
                int pos = TOPK - 1;
                #pragma unroll
                for (int s = TOPK - 1; s > 0; --s) {
                    if (val[s - 1] < v) {
                        val[s] = val[s - 1];
                        idx[s] = idx[s - 1];
                        pos = s - 1;
                    }
                }
                val[pos] = v;
                idx[pos] = e;
            }
        }

        // softmax over the 8 selected logits (val[0] is the max)
        float m = val[0];
        float sum = 0.f;
        float ex[TOPK];
        #pragma unroll
        for (int i = 0; i < TOPK; ++i) { ex[i] = __expf(val[i] - m); sum += ex[i]; }
        float inv = 1.f / sum;

        const int token = t0 + tid;
        #pragma unroll
        for (int i = 0; i < TOPK; ++i) {
            probs_out[(size_t)token * TOPK + i] = ex[i] * inv;
            idx_out[(size_t)token * TOPK + i]   = idx[i];
        }
    }
}

extern "C" void kernel_launch(void* const* d_in, const int* in_sizes, int n_in,
                              void* d_out, int out_size, void* d_ws, size_t ws_size,
                              hipStream_t stream) {
    (void)n_in; (void)d_ws; (void)ws_size; (void)out_size;
    const float* input  = (const float*)d_in[0];
    const float* weight = (const float*)d_in[1];

    const int num_tokens = in_sizes[0] / HID;   // 16384

    float* probs = (float*)d_out;                                   // first T*8 floats
    int*   idx   = (int*)d_out + (size_t)num_tokens * TOPK;         // next T*8 int32

    dim3 grid(num_tokens / MT), block(128);
    topk_router_kernel<<<grid, block, 0, stream>>>(input, weight, probs, idx, num_tokens);
}